// ModelNew_25056839205233
// MI455X (gfx1250) — compile-verified
//
#include <hip/hip_runtime.h>
#include <stdint.h>

typedef uint32_t u32;
typedef uint64_t u64;
typedef __attribute__((ext_vector_type(4))) u32   v4u;
typedef __attribute__((ext_vector_type(8))) u32   v8u;
typedef __attribute__((ext_vector_type(4))) float v4f;

#define HW    49      // H == W == 49
#define DTOT  512
#define DSLAB 256     // channels per block (LDS tile = 256*49*4 = 50,176 B)

__global__ __launch_bounds__(256)
void patchify_tdm_kernel(const float* __restrict__ x, float* __restrict__ out) {
    __shared__ float tile[DSLAB * HW];   // tile[d_local*49 + w] = x[b, d0+d_local, h, w]

    const int slab = blockIdx.x;   // 0..1
    const int h    = blockIdx.y;   // 0..48
    const int b    = blockIdx.z;   // 0..63
    const int d0   = slab * DSLAB;

    // ---- TDM: DMA the strided 256x49 f32 tile into LDS (one issue per block) ----
    if (threadIdx.x == 0) {
        // byte address of x[b, d0, h, 0]; rows advance by 49*49 elements per channel
        u64 gaddr = (u64)(uintptr_t)(x + (((u64)(b * DTOT + d0) * HW + (u64)h) * HW));
        // low 32 bits of a generic pointer to LDS == workgroup-relative LDS byte offset
        u32 ldsa  = (u32)(uintptr_t)(&tile[0]);

        v4u g0;
        g0.x = 1u;                                              // count=1, is_restore=0, gather off
        g0.y = ldsa;                                            // lds_addr [63:32]
        g0.z = (u32)(gaddr & 0xFFFFFFFFu);                      // global_addr[31:0]
        g0.w = (u32)((gaddr >> 32) & 0x01FFFFFFu) | (2u << 30); // global_addr[56:32], type=2

        v8u g1;
        g1.s0 = (2u << 16);            // workgroup_mask=0, data_size=2 (4 bytes)
        g1.s1 = ((u32)HW << 16);       // tensor_dim0[15:0] = 49   (bits 63:48)
        g1.s2 = ((u32)DSLAB << 16);    // tensor_dim0[31:16]=0, tensor_dim1[15:0]=256
        g1.s3 = ((u32)HW << 16);       // tensor_dim1[31:16]=0, tile_dim0=49 (bits 127:112)
        g1.s4 = (u32)DSLAB;            // tile_dim1=256 (bits 143:128), tile_dim2=0
        g1.s5 = (u32)(HW * HW);        // tensor_dim0_stride[31:0] = 2401 elements
        g1.s6 = 0u;                    // stride0[47:32]=0, tensor_dim1_stride[15:0]=0
        g1.s7 = 0u;

        // 2D tensor: descriptor groups 2/3 are NULL -> 2-operand form
        asm volatile("tensor_load_to_lds %0, %1" :: "s"(g0), "s"(g1) : "memory");
        __builtin_amdgcn_s_wait_tensorcnt(0);
    }
    __syncthreads();

    // ---- transpose out of LDS; 512B-contiguous NT stores per wave ----
    const int ih  = h / 7;
    const int iph = h % 7;
    float* outb = out + (size_t)b * (HW * HW) * DTOT + d0;

    // 64 float4-chunks of D per w; lanes of a wave share w -> coalesced 512B stores
    for (int idx = threadIdx.x; idx < HW * (DSLAB / 4); idx += 256) {
        const int w   = idx >> 6;        // 0..48 (uniform within a wave)
        const int j   = idx & 63;        // d-quad index
        const int iw  = w / 7;
        const int ipw = w % 7;
        const int p   = iph * 7 + ipw;   // intra-patch position
        const int t   = ih  * 7 + iw;    // patch/token index
        const int dl  = j * 4;

        v4f v;
        v.x = tile[(dl + 0) * HW + w];
        v.y = tile[(dl + 1) * HW + w];
        v.z = tile[(dl + 2) * HW + w];
        v.w = tile[(dl + 3) * HW + w];

        v4f* dst = reinterpret_cast<v4f*>(outb + (size_t)(p * HW + t) * DTOT + dl);
        __builtin_nontemporal_store(v, dst);
    }
}

extern "C" void kernel_launch(void* const* d_in, const int* in_sizes, int n_in,
                              void* d_out, int out_size, void* d_ws, size_t ws_size,
                              hipStream_t stream) {
    (void)in_sizes; (void)n_in; (void)out_size; (void)d_ws; (void)ws_size;
    const float* x = (const float*)d_in[0];
    float* out = (float*)d_out;

    dim3 grid(2, HW, 64);   // (d-slab, h, b) ; b outermost for L2 locality
    dim3 block(256);
    hipLaunchKernelGGL(patchify_tdm_kernel, grid, block, 0, stream, x, out);
}